// NeuralVMEmbedding_81020263072229
// MI455X (gfx1250) — compile-verified
//
#include <hip/hip_runtime.h>

// ---- problem constants (from reference) ----
#define NVM_VOCAB 272
#define NVM_D     512
#define NVM_B     8
#define NVM_S     8192
#define TOK_CODE_START 256
#define TOK_CODE_END   257
#define TOK_MEM        258
#define TOK_THINK_S    259
#define TOK_THINK_E    260
// ADDR_KEY=206 (48 feats), MARK_TS=456, MARK_TE=457, MEM_EXEC=458

typedef unsigned int u32;
typedef u32 u32x4 __attribute__((ext_vector_type(4)));
typedef u32 u32x8 __attribute__((ext_vector_type(8)));
typedef float f32x4 __attribute__((ext_vector_type(4)));

// ============================================================================
// Kernel A: per-row scan.  cs_pos[b,s] = cummax_{j<=s}(tok[j]==CODE_START ? j : -1)
//           first_ce[b]  = min{ s : tok[s]==CODE_END } else S
// One block (1024 threads) per row; each thread owns 8 contiguous tokens.
// ============================================================================
__global__ __launch_bounds__(1024) void nvm_scan(const int* __restrict__ tok,
                                                 int* __restrict__ cs_pos,
                                                 int* __restrict__ first_ce) {
  __shared__ int sm[1024];
  const int b   = blockIdx.x;
  const int tid = threadIdx.x;
  const int* t  = tok + (size_t)b * NVM_S;
  const int base = tid * 8;

  int lmax = -1, lmin = NVM_S;
#pragma unroll
  for (int k = 0; k < 8; ++k) {
    const int v = t[base + k];
    if (v == TOK_CODE_START) lmax = base + k;               // increasing k => max
    if (v == TOK_CODE_END && (base + k) < lmin) lmin = base + k;
  }

  // inclusive max-scan over per-thread maxima (Hillis-Steele)
  sm[tid] = lmax;
  __syncthreads();
  for (int off = 1; off < 1024; off <<= 1) {
    int v = sm[tid];
    if (tid >= off) v = max(v, sm[tid - off]);
    __syncthreads();
    sm[tid] = v;
    __syncthreads();
  }
  int run = (tid > 0) ? sm[tid - 1] : -1;                    // exclusive prefix
#pragma unroll
  for (int k = 0; k < 8; ++k) {
    const int v = t[base + k];
    if (v == TOK_CODE_START) run = base + k;
    cs_pos[(size_t)b * NVM_S + base + k] = run;
  }

  // min-reduce for first CODE_END
  __syncthreads();
  sm[tid] = lmin;
  __syncthreads();
  for (int off = 512; off > 0; off >>= 1) {
    if (tid < off) sm[tid] = min(sm[tid], sm[tid + off]);
    __syncthreads();
  }
  if (tid == 0) first_ce[b] = sm[0];
}

// ============================================================================
// Kernel B: gather + overrides.  8 waves/block, one wave per token.
//  - token window [s0-8, s0+8) staged via TDM tensor_load_to_lds (TENSORcnt)
//  - cs_pos window staged via global_load_async_to_lds_b32 (ASYNCcnt)
//  - output written with non-temporal B128 stores (keep embed table in L2)
// ============================================================================
__device__ __forceinline__ void set_oh48(unsigned long long& oh, int addr) {
  oh |= (1ull << (addr & 15))
      | (1ull << (16 + ((addr >> 4) & 15)))
      | (1ull << (32 + ((addr >> 8) & 15)));
}

__global__ __launch_bounds__(256) void nvm_main(const float* __restrict__ embed,
                                                const int* __restrict__ tok,
                                                const int* __restrict__ cs_pos,
                                                const int* __restrict__ first_ce,
                                                float* __restrict__ out) {
  __shared__ int stok[24];   // token window landing zone (shift-adjusted)
  __shared__ int scsp[8];    // cs_pos for the block's 8 tokens

  const int g0 = blockIdx.x * 8;          // first flattened token of block
  const int b  = g0 / NVM_S;
  const int s0 = g0 % NVM_S;              // multiple of 8 -> block stays in one row

  // --- TDM: stage 16 tokens starting at flat index (g0-8), clamped at 0.
  // stok[rel] corresponds to row-local position (s0 - 8 + rel); out-of-row
  // entries are never consumed (guarded by row-local i >= 0 checks below).
  if (threadIdx.x < 32) {                 // wave 0 issues the tensor op
    const int wstart = g0 - 8;
    const int wclamp = (wstart < 0) ? 0 : wstart;     // only block 0 clamps
    const unsigned shift = (unsigned)(wclamp - wstart); // 0 or 8
    const unsigned long long ga = (unsigned long long)(const void*)(tok + wclamp);
    const unsigned lds0 = (unsigned)(size_t)(&stok[0]) + shift * 4u;

    u32x4 d0;                      // D# group 0
    d0[0] = 1u;                                        // count=1, no gather
    d0[1] = lds0;                                      // lds_addr
    d0[2] = (u32)ga;                                   // global_addr[31:0]
    d0[3] = ((u32)(ga >> 32) & 0x01FFFFFFu) | (2u << 30); // ga[56:32] | type=2

    u32x8 d1;                      // D# group 1
    d1[0] = 2u << 16;              // workgroup_mask=0, data_size=2 (4 bytes)
    d1[1] = 16u << 16;             // tensor_dim0[15:0] = 16
    d1[2] = 1u << 16;              // tensor_dim0[31:16]=0, tensor_dim1[15:0]=1
    d1[3] = 16u << 16;             // tensor_dim1[31:16]=0, tile_dim0 = 16
    d1[4] = 0u;                    // tile_dim1=0 (unused), tile_dim2=0
    d1[5] = 16u;                   // tensor_dim0_stride[31:0] = 16
    d1[6] = 0u;
    d1[7] = 0u;
    asm volatile("tensor_load_to_lds %0, %1" :: "s"(d0), "s"(d1) : "memory");
  }

  // --- async-to-LDS: stage the 8 cs_pos values for this block
  if (threadIdx.x < 8) {
    const int* gp = cs_pos + g0 + (int)threadIdx.x;
    const unsigned loff = (unsigned)(size_t)(&scsp[threadIdx.x]);
    asm volatile("global_load_async_to_lds_b32 %0, %1, off"
                 :: "v"(loff), "v"(gp) : "memory");
  }

  __builtin_amdgcn_s_wait_tensorcnt(0);
  asm volatile("s_wait_asynccnt 0" ::: "memory");
  __syncthreads();

  const int wave = threadIdx.x >> 5;
  const int lane = threadIdx.x & 31;
  const int s = s0 + wave;                 // row-local position
  const int g = g0 + wave;                 // flattened token index

  const int t   = stok[s - s0 + 8];
  const int cs  = scsp[wave];
  const int fce = first_ce[b];

  // ---- build oh48 bitmask ----
  unsigned long long oh = 0;
  // code path
  if (cs >= 0 && s < fce && t < 256) {
    const int sp = s - cs - 1;
    if (sp >= 0) {
      const int bo = sp & 7;               // seq_pos % BYTES_PER_INSTR
      if (bo < 5) {                        // byte_off < DATA_BYTES
        const int caddr = ((sp & ~7) + 2 + bo) & 4095; // (sp/8)*8 + PC_OFFSET + bo
        set_oh48(oh, caddr);
      }
    }
  }
  // MEM look-back path: off in 0..3, source position i = s - 5 - off
#pragma unroll
  for (int off = 0; off < 4; ++off) {
    const int i = s - 5 - off;
    if (i >= 0) {
      const int rel = i - s0 + 8;          // in [0,11)
      if (stok[rel] == TOK_MEM && (i + 8) < NVM_S) {
        const int a12 = (stok[rel + 1] | (stok[rel + 2] << 8)) & 4095;
        set_oh48(oh, (a12 + off) & 4095);
      }
    }
  }
  const bool memm = (t == TOK_MEM) && (s + 8 < NVM_S);

  // ---- move one 512-float row: lane l handles float4 indices l, l+32, l+64, l+96
  const f32x4* __restrict__ erow = (const f32x4*)(embed + (size_t)t * NVM_D);
  f32x4* __restrict__ orow       = (f32x4*)(out + (size_t)g * NVM_D);

  // quarter 0: features [0,128)   — pure copy (NT streaming store)
  __builtin_nontemporal_store(erow[lane], &orow[lane]);
  // quarter 2: features [256,384) — pure copy
  __builtin_nontemporal_store(erow[lane + 64], &orow[lane + 64]);

  // quarter 1: features [128,256) — apply oh48 overrides on [206,254)
  {
    f32x4 v = erow[lane + 32];
    const int d0 = 128 + 4 * lane;
#pragma unroll
    for (int j = 0; j < 4; ++j) {
      const unsigned k = (unsigned)(d0 + j - 206);
      if (k < 48u && ((oh >> k) & 1ull)) v[j] = 1.0f;
    }
    __builtin_nontemporal_store(v, &orow[lane + 32]);
  }

  // quarter 3: features [384,512) — flags at 456/457/458
  {
    f32x4 v = erow[lane + 96];
    const int d0 = 384 + 4 * lane;
#pragma unroll
    for (int j = 0; j < 4; ++j) {
      const int d = d0 + j;
      if (d == 456 && t == TOK_THINK_S) v[j] = 1.0f;
      if (d == 457 && t == TOK_THINK_E) v[j] = 1.0f;
      if (d == 458 && memm)             v[j] = 1.0f;
    }
    __builtin_nontemporal_store(v, &orow[lane + 96]);
  }
}

// ============================================================================
extern "C" void kernel_launch(void* const* d_in, const int* in_sizes, int n_in,
                              void* d_out, int out_size, void* d_ws, size_t ws_size,
                              hipStream_t stream) {
  const float* embed = (const float*)d_in[0];   // [272,512] f32
  const int*   tokid = (const int*)d_in[1];     // [8,8192]  i32
  float* out = (float*)d_out;                   // [8,8192,512] f32

  int* cs_pos   = (int*)d_ws;                   // B*S ints
  int* first_ce = cs_pos + NVM_B * NVM_S;       // B ints

  nvm_scan<<<NVM_B, 1024, 0, stream>>>(tokid, cs_pos, first_ce);
  nvm_main<<<(NVM_B * NVM_S) / 8, 256, 0, stream>>>(embed, tokid, cs_pos, first_ce, out);
}